// HadamardRotation_8890582303360
// MI455X (gfx1250) — compile-verified
//
#include <hip/hip_runtime.h>

// FWHT-4096 via H16 (x) H16 (x) H16 Kronecker factorization, each 16x16x16
// product done with 4 chained V_WMMA_F32_16X16X4_F32. One wave32 per row.

typedef __attribute__((ext_vector_type(2))) float v2f;
typedef __attribute__((ext_vector_type(8))) float v8f;

#define WAVES 2
#define SA 332                 // LDS stride between 'a' groups (floats); 332 % 8 == 4
#define SB 20                  // LDS stride between 'b' rows (floats)
#define LDS_ROW (16 * SA)      // 5312 floats per row

__device__ __forceinline__ float hsign(int i, int j) {
    return (__popc(i & j) & 1) ? -1.0f : 1.0f;
}

__device__ __forceinline__ v8f wmma4(v2f a, v2f b, v8f c) {
    // (neg_a, A, neg_b, B, c_mod, C, reuse_a, reuse_b)
    return __builtin_amdgcn_wmma_f32_16x16x4_f32(false, a, false, b, (short)0, c,
                                                 false, false);
}

__global__ __launch_bounds__(WAVES * 32)
void fwht4096_wmma_kernel(const float* __restrict__ x, float* __restrict__ out,
                          int nrows) {
    __shared__ float lds[WAVES * LDS_ROW];
    const int lane = threadIdx.x & 31;
    const int wave = threadIdx.x >> 5;
    const int row  = blockIdx.x * WAVES + wave;
    if (row >= nrows) return;          // uniform per wave -> EXEC stays all-1s

    float* L = lds + wave * LDS_ROW;
    const float* g = x + (size_t)row * 4096;

    // ---- Stage row into LDS (coalesced b128 loads, padded b128 stores) ----
    for (int t = lane; t < 1024; t += 32) {
        float4 v = reinterpret_cast<const float4*>(g)[t];
        int i = t << 2;
        int a = i >> 8;
        int b = (i >> 4) & 15;
        int c = i & 15;                 // c in {0,4,8,12} -> 16B aligned in LDS
        *reinterpret_cast<float4*>(L + a * SA + b * SB + c) = v;
    }

    const int half = lane >> 4;         // 0|1
    const int lo   = lane & 15;

    // H16 operand, chunked for K=4 WMMAs. Symmetric, so it serves as the B
    // operand in pass 1 and the A operand in passes 2 & 3 (same lane layout:
    // value index k = 4*kc + j + 2*half paired with m/n = lo).
    v2f hk[4];
#pragma unroll
    for (int kc = 0; kc < 4; ++kc) {
        int k = 4 * kc + 2 * half;
        hk[kc].x = hsign(k + 0, lo);
        hk[kc].y = hsign(k + 1, lo);
    }

    // ---- Pass 1: along c.  Y_a = X_a * H  (A = data, B = H) ----
    for (int a = 0; a < 16; ++a) {
        const float* base = L + a * SA;
        v8f acc = {};
#pragma unroll
        for (int kc = 0; kc < 4; ++kc) {
            int k = 4 * kc + 2 * half;
            v2f av;
            av.x = base[lo * SB + k + 0];   // A[m=lo][k]
            av.y = base[lo * SB + k + 1];
            acc = wmma4(av, hk[kc], acc);
        }
        float* ob = L + a * SA + (8 * half) * SB + lo;  // D[i+8*half][lo]
#pragma unroll
        for (int i = 0; i < 8; ++i) ob[i * SB] = acc[i];
    }

    // ---- Pass 2: along b.  Z_a = H * Y_a  (A = H, B = data) ----
    for (int a = 0; a < 16; ++a) {
        const float* base = L + a * SA;
        v8f acc = {};
#pragma unroll
        for (int kc = 0; kc < 4; ++kc) {
            int k = 4 * kc + 2 * half;
            v2f bv;
            bv.x = base[(k + 0) * SB + lo]; // B[k][n=lo]
            bv.y = base[(k + 1) * SB + lo];
            acc = wmma4(hk[kc], bv, acc);
        }
        float* ob = L + a * SA + (8 * half) * SB + lo;
#pragma unroll
        for (int i = 0; i < 8; ++i) ob[i * SB] = acc[i];
    }

    // ---- Pass 3: along a.  W = H * Z, scaled by 1/64, straight to global ----
    float* o = out + (size_t)row * 4096;
    const float scale = 0.015625f;      // 1/sqrt(4096)
    for (int gci = 0; gci < 16; ++gci) {            // gci = fixed 's' index
        v8f acc = {};
#pragma unroll
        for (int kc = 0; kc < 4; ++kc) {
            int k = 4 * kc + 2 * half;
            v2f bv;
            bv.x = L[(k + 0) * SA + gci * SB + lo]; // B[k=a][n=lo]
            bv.y = L[(k + 1) * SA + gci * SB + lo];
            acc = wmma4(hk[kc], bv, acc);
        }
        // D row r = i + 8*half, out index r*256 + gci*16 + lo
        float* ob = o + (8 * half) * 256 + gci * 16 + lo;
#pragma unroll
        for (int i = 0; i < 8; ++i) ob[i * 256] = acc[i] * scale;
    }
}

extern "C" void kernel_launch(void* const* d_in, const int* in_sizes, int n_in,
                              void* d_out, int out_size, void* d_ws, size_t ws_size,
                              hipStream_t stream) {
    const float* x = (const float*)d_in[0];
    // d_in[1] (dense Hadamard matrix) intentionally unused: H entries are
    // regenerated analytically as (-1)^popcount(i&j), exactly matching the
    // Sylvester construction in the reference.
    float* out = (float*)d_out;

    int nrows  = in_sizes[0] / 4096;                 // 16384 for [4,4096,4096]
    int blocks = (nrows + WAVES - 1) / WAVES;
    fwht4096_wmma_kernel<<<blocks, WAVES * 32, 0, stream>>>(x, out, nrows);
}